// HyFMConv_24635932410412
// MI455X (gfx1250) — compile-verified
//
#include <hip/hip_runtime.h>

typedef float v2f __attribute__((ext_vector_type(2)));
typedef float v8f __attribute__((ext_vector_type(8)));

constexpr int N_NODES   = 100000;
constexpr int N_HYEDGES = 20000;
constexpr int NNZ       = 1600000;
constexpr int IN_CH     = 256;
constexpr int OUT_CH    = 128;

// ---------------------------------------------------------------------------
// GEMM: y[N,128] = x[N,256] @ theta[256,128] in exact f32 via
// V_WMMA_F32_16X16X4_F32. One wave32 owns a 16x128 output strip: 8 col-tiles
// of 16x16 f32 accumulators (64 VGPRs), A fragment reused across all 8 WMMAs
// per k-step. Memory-bound (~42 FLOP/B), so f32 WMMA == peak here.
// ---------------------------------------------------------------------------
__global__ __launch_bounds__(256) void hyfm_gemm_wmma_f32(
    const float* __restrict__ x, const float* __restrict__ theta,
    float* __restrict__ y)
{
    const int wave    = blockIdx.x * (blockDim.x >> 5) + (threadIdx.x >> 5);
    const int n_tiles = N_NODES / 16;           // 6250, exact
    if (wave >= n_tiles) return;                // whole-wave uniform exit
    const int lane = threadIdx.x & 31;
    const int r    = lane & 15;                 // M for A, N for B/C
    const int half = lane >> 4;                 // K-half selector

    const float* __restrict__ xrow = x + (size_t)(wave * 16 + r) * IN_CH;

    v8f acc[8] = {};                            // 8 col tiles x 8 VGPRs

    for (int kk = 0; kk < IN_CH; kk += 4) {
        const int k0 = kk + 2 * half;
        // A 16x4 f32 layout: lane r, VGPR0=K(k0), VGPR1=K(k0+1) -> v2f load
        v2f a = *(const v2f*)(xrow + k0);
        const float* __restrict__ th0 = theta + (size_t)k0 * OUT_CH + r;
#pragma unroll
        for (int c = 0; c < 8; ++c) {
            // B 4x16 f32 layout: N=r across lanes, v0=K(k0), v1=K(k0+1)
            v2f b;
            b.x = th0[c * 16];
            b.y = th0[c * 16 + OUT_CH];
            acc[c] = __builtin_amdgcn_wmma_f32_16x16x4_f32(
                false, a, false, b, (short)0, acc[c], false, false);
        }
    }

    // C/D layout: VGPR i -> M = i + 8*half, N = r
#pragma unroll
    for (int c = 0; c < 8; ++c) {
#pragma unroll
        for (int i = 0; i < 8; ++i) {
            y[(size_t)(wave * 16 + i + 8 * half) * OUT_CH + c * 16 + r] = acc[c][i];
        }
    }
}

// ---------------------------------------------------------------------------
// Degree histograms (float atomics straight into L2-resident arrays)
// ---------------------------------------------------------------------------
__global__ __launch_bounds__(256) void hyfm_degree(
    const int* __restrict__ nid, const int* __restrict__ eid,
    float* __restrict__ deg_n, float* __restrict__ deg_e, int nnz)
{
    int i = blockIdx.x * blockDim.x + threadIdx.x;
    if (i >= nnz) return;
    atomicAdd(&deg_e[eid[i]], 1.0f);
    atomicAdd(&deg_n[nid[i]], 1.0f);
}

__global__ __launch_bounds__(256) void hyfm_recip(float* __restrict__ d, int n)
{
    int i = blockIdx.x * blockDim.x + threadIdx.x;
    if (i >= n) return;
    float v = d[i];
    d[i] = (v > 0.0f) ? (1.0f / v) : 0.0f;
}

__global__ __launch_bounds__(256) void hyfm_bias_init(
    const float* __restrict__ bias, float* __restrict__ out, int total)
{
    int i = blockIdx.x * blockDim.x + threadIdx.x;
    if (i >= total) return;
    out[i] = bias[i & (OUT_CH - 1)];
}

// ---------------------------------------------------------------------------
// Scatter stage: dst[sidx[j]] += src[gidx[j]] * inv_deg[sidx[j]]
// One wave per nnz entry, each lane moves 4 channels (float4 gather + 4
// non-returning global_atomic_add_f32). 512B-contiguous rows keep HBM happy;
// e_ft (10 MB) / d_out (51 MB) accumulate inside the 192 MB L2.
// ---------------------------------------------------------------------------
__global__ __launch_bounds__(256) void hyfm_scatter(
    const float* __restrict__ src, float* __restrict__ dst,
    const int* __restrict__ gidx, const int* __restrict__ sidx,
    const float* __restrict__ inv_deg, int nnz)
{
    const int w = blockIdx.x * (blockDim.x >> 5) + (threadIdx.x >> 5);
    if (w >= nnz) return;
    const int lane = threadIdx.x & 31;
    const int g = gidx[w];
    const int s = sidx[w];
    const float sc = inv_deg[s];
    const float4 v = ((const float4*)(src + (size_t)g * OUT_CH))[lane];
    float* drow = dst + (size_t)s * OUT_CH + lane * 4;
    atomicAdd(drow + 0, v.x * sc);
    atomicAdd(drow + 1, v.y * sc);
    atomicAdd(drow + 2, v.z * sc);
    atomicAdd(drow + 3, v.w * sc);
}

// ---------------------------------------------------------------------------
extern "C" void kernel_launch(void* const* d_in, const int* in_sizes, int n_in,
                              void* d_out, int out_size, void* d_ws, size_t ws_size,
                              hipStream_t stream)
{
    const float* x          = (const float*)d_in[0];
    const float* theta      = (const float*)d_in[1];
    const float* bias       = (const float*)d_in[2];
    const int*   node_idx   = (const int*)d_in[3];
    const int*   hyedge_idx = (const int*)d_in[4];
    float*       out        = (float*)d_out;

    // Workspace layout (bytes):
    //   [0,            51,200,000)  y     = x @ theta
    //   [51,200,000,   61,440,000)  e_ft  (zeroed each call)
    //   [61,440,000,   61,520,000)  deg_e (zeroed each call)
    //   [61,520,000,   61,920,000)  deg_n (zeroed each call)
    char*  ws    = (char*)d_ws;
    float* y     = (float*)(ws);
    float* e_ft  = (float*)(ws + 51200000);
    float* deg_e = (float*)(ws + 61440000);
    float* deg_n = (float*)(ws + 61520000);

    // zero e_ft + both degree arrays in one contiguous async memset
    hipMemsetAsync(ws + 51200000, 0, 10720000, stream);

    hyfm_degree<<<(NNZ + 255) / 256, 256, 0, stream>>>(
        node_idx, hyedge_idx, deg_n, deg_e, NNZ);
    // deg_e and deg_n are contiguous: one reciprocal pass over both
    hyfm_recip<<<(N_HYEDGES + N_NODES + 255) / 256, 256, 0, stream>>>(
        deg_e, N_HYEDGES + N_NODES);

    // projection GEMM: 6250 row-tiles, 8 waves per 256-thread block
    hyfm_gemm_wmma_f32<<<(N_NODES / 16 + 7) / 8, 256, 0, stream>>>(x, theta, y);

    // out = bias (broadcast), then scatter-add hyperedge->node directly into it
    hyfm_bias_init<<<(N_NODES * OUT_CH + 255) / 256, 256, 0, stream>>>(
        bias, out, N_NODES * OUT_CH);

    // node -> hyperedge
    hyfm_scatter<<<(NNZ + 7) / 8, 256, 0, stream>>>(
        y, e_ft, node_idx, hyedge_idx, deg_e, NNZ);
    // hyperedge -> node (accumulates onto bias-initialized out)
    hyfm_scatter<<<(NNZ + 7) / 8, 256, 0, stream>>>(
        e_ft, out, hyedge_idx, node_idx, deg_n, NNZ);
}